// DCTChannelBlock_25400436589285
// MI455X (gfx1250) — compile-verified
//
#include <hip/hip_runtime.h>
#include <hip/hip_bf16.h>

typedef __attribute__((ext_vector_type(2))) float v2f;
typedef __attribute__((ext_vector_type(8))) float v8f;

#define B_SZ   32
#define C_SZ   512
#define L_SZ   1024
#define ROWS   (B_SZ * C_SZ)      // 16384
#define TWO_C  (2 * C_SZ)         // 1024

// ---------------------------------------------------------------------------
// DCT matrix generation: DT[n][k] = 2*cos(pi*k*(2n+1)/(2L)), stored [n][k]
// so that freq[r,k] = sum_n X[r,n] * DT[n][k]  (B operand, row-major, ldb=L).
// Exact integer phase reduction (k*(2n+1) mod 4096) keeps the fp32 hw cosine
// argument inside one revolution -> full fp32 accuracy, no fp64 needed.
// ---------------------------------------------------------------------------
__global__ void dct_gen_kernel(float* __restrict__ DT) {
    const int idx = blockIdx.x * 256 + threadIdx.x;   // < 1024*1024
    const int n = idx >> 10;
    const int k = idx & 1023;
    const int m = (k * (2 * n + 1)) & 4095;           // phase mod 2*pi
    DT[idx] = 2.0f * __cosf((float)(3.14159265358979323846 / 2048.0) * (float)m);
}

// ---------------------------------------------------------------------------
// Wave-level register-blocked fp32 WMMA GEMM: each wave computes a 64x32
// output tile as 4x2 WMMA 16x16 tiles. Per K-step of 4: 4 A-frag loads +
// 2 B-frag loads feed 8 v_wmma_f32_16x16x4_f32 (1.0 vmem/wmma, A reused 2x,
// B reused 4x from registers; 8 independent accumulators hide the
// WMMA->WMMA RAW hazard window).
// A[M,K] row-major (lda), B[K,N] row-major (ldb).
// A frag (16x4 f32): lanes 0-15 -> M=lane, vgpr0/1 = K=0/1;
//                    lanes16-31 -> M=lane-16, vgpr0/1 = K=2/3.
// B frag (4x16 f32): mirrored with N in place of M.
// C/D: lane 0-15 -> N=lane, vgpr r = M=r; lanes 16-31 -> M=r+8.
// ---------------------------------------------------------------------------
__device__ __forceinline__ void gemm_tile_64x32(const float* __restrict__ A, int lda,
                                                const float* __restrict__ B, int ldb,
                                                int tileM, int tileN, int K,
                                                v8f acc[4][2]) {
    const int lane = threadIdx.x & 31;
    const int half = lane >> 4;
    const int mn   = lane & 15;
    const float* __restrict__ Ar[4];
#pragma unroll
    for (int i = 0; i < 4; ++i)
        Ar[i] = A + (size_t)(tileM + 16 * i + mn) * lda;
    const float* __restrict__ Bc0 = B + (tileN + mn);
    const float* __restrict__ Bc1 = Bc0 + 16;

    for (int k = 0; k < K; k += 8) {
#pragma unroll
        for (int kk = 0; kk < 8; kk += 4) {
            const int kb = k + kk + 2 * half;
            v2f a[4], b0, b1;
#pragma unroll
            for (int i = 0; i < 4; ++i) {
                a[i].x = Ar[i][kb];
                a[i].y = Ar[i][kb + 1];
            }
            const size_t o0 = (size_t)kb * ldb;
            const size_t o1 = o0 + ldb;
            b0.x = Bc0[o0]; b0.y = Bc0[o1];
            b1.x = Bc1[o0]; b1.y = Bc1[o1];
#pragma unroll
            for (int i = 0; i < 4; ++i) {
                acc[i][0] = __builtin_amdgcn_wmma_f32_16x16x4_f32(
                    false, a[i], false, b0, (short)0, acc[i][0], false, false);
                acc[i][1] = __builtin_amdgcn_wmma_f32_16x16x4_f32(
                    false, a[i], false, b1, (short)0, acc[i][1], false, false);
            }
        }
    }
}

// ---------------------------------------------------------------------------
// GEMM 1: F[ROWS x L] = X[ROWS x L] * DT[L x L]
// 8 waves per block arranged 2(M) x 4(N): block tile = 128x128.
// ---------------------------------------------------------------------------
__global__ void dct_gemm_kernel(const float* __restrict__ X,
                                const float* __restrict__ DT,
                                float* __restrict__ F) {
    const int wid   = threadIdx.x >> 5;
    const int bm    = blockIdx.x >> 3;                 // 1024/128 = 8 N-blocks
    const int bn    = blockIdx.x & 7;
    const int tileM = bm * 128 + (wid >> 2) * 64;
    const int tileN = bn * 128 + (wid & 3) * 32;

    v8f acc[4][2] = {};
    gemm_tile_64x32(X, L_SZ, DT, L_SZ, tileM, tileN, L_SZ, acc);

    const int lane = threadIdx.x & 31;
    const int half = lane >> 4;
    const int n    = lane & 15;
#pragma unroll
    for (int i = 0; i < 4; ++i)
#pragma unroll
        for (int j = 0; j < 2; ++j)
#pragma unroll
            for (int r = 0; r < 8; ++r)
                F[(size_t)(tileM + 16 * i + r + 8 * half) * L_SZ +
                  tileN + 16 * j + n] = acc[i][j][r];
}

// ---------------------------------------------------------------------------
// LayerNorm over each length-1024 row of F, in place. One block per row.
// ---------------------------------------------------------------------------
__global__ void layernorm_kernel(float* __restrict__ F,
                                 const float* __restrict__ lnw,
                                 const float* __restrict__ lnb) {
    __shared__ float s1[256];
    __shared__ float s2[256];
    float* __restrict__ row = F + (size_t)blockIdx.x * L_SZ;
    const int tid = threadIdx.x;

    float v[4];
    float sum = 0.f, sq = 0.f;
#pragma unroll
    for (int j = 0; j < 4; ++j) {
        v[j] = row[tid + 256 * j];
        sum += v[j];
        sq  += v[j] * v[j];
    }
    s1[tid] = sum;
    s2[tid] = sq;
    __syncthreads();
    for (int off = 128; off > 0; off >>= 1) {
        if (tid < off) {
            s1[tid] += s1[tid + off];
            s2[tid] += s2[tid + off];
        }
        __syncthreads();
    }
    const float mean = s1[0] * (1.f / 1024.f);
    const float var  = s2[0] * (1.f / 1024.f) - mean * mean;
    const float rstd = rsqrtf(var + 1e-6f);
#pragma unroll
    for (int j = 0; j < 4; ++j) {
        const int idx = tid + 256 * j;
        row[idx] = (v[j] - mean) * rstd * lnw[idx] + lnb[idx];
    }
}

// ---------------------------------------------------------------------------
// GEMM 2 (+ReLU): per batch b: H1_b[2c x L] = relu(W1[2c x c] * H0_b[c x L])
// ---------------------------------------------------------------------------
__global__ void fc1_relu_kernel(const float* __restrict__ W1,
                                const float* __restrict__ H0,
                                float* __restrict__ H1) {
    const int b     = blockIdx.y;
    const int wid   = threadIdx.x >> 5;
    const int bm    = blockIdx.x >> 3;                 // M=1024 -> 8 blocks
    const int bn    = blockIdx.x & 7;
    const int tileM = bm * 128 + (wid >> 2) * 64;
    const int tileN = bn * 128 + (wid & 3) * 32;

    const float* __restrict__ H0b = H0 + (size_t)b * C_SZ * L_SZ;
    float* __restrict__ H1b       = H1 + (size_t)b * TWO_C * L_SZ;

    v8f acc[4][2] = {};
    gemm_tile_64x32(W1, C_SZ, H0b, L_SZ, tileM, tileN, C_SZ, acc);

    const int lane = threadIdx.x & 31;
    const int half = lane >> 4;
    const int n    = lane & 15;
#pragma unroll
    for (int i = 0; i < 4; ++i)
#pragma unroll
        for (int j = 0; j < 2; ++j)
#pragma unroll
            for (int r = 0; r < 8; ++r)
                H1b[(size_t)(tileM + 16 * i + r + 8 * half) * L_SZ +
                    tileN + 16 * j + n] = fmaxf(acc[i][j][r], 0.f);
}

// ---------------------------------------------------------------------------
// GEMM 3 (+sigmoid gate): out[b,o,l] = x[b,o,l] * sigmoid(sum_i W2[o,i]*H1[b,i,l])
// ---------------------------------------------------------------------------
__global__ void fc2_gate_kernel(const float* __restrict__ W2,
                                const float* __restrict__ H1,
                                const float* __restrict__ X,
                                float* __restrict__ out) {
    const int b     = blockIdx.y;
    const int wid   = threadIdx.x >> 5;
    const int bm    = blockIdx.x >> 3;                 // M=512 -> 4 blocks
    const int bn    = blockIdx.x & 7;
    const int tileM = bm * 128 + (wid >> 2) * 64;
    const int tileN = bn * 128 + (wid & 3) * 32;

    const float* __restrict__ H1b = H1 + (size_t)b * TWO_C * L_SZ;

    v8f acc[4][2] = {};
    gemm_tile_64x32(W2, TWO_C, H1b, L_SZ, tileM, tileN, TWO_C, acc);

    const int lane = threadIdx.x & 31;
    const int half = lane >> 4;
    const int n    = lane & 15;
#pragma unroll
    for (int i = 0; i < 4; ++i)
#pragma unroll
        for (int j = 0; j < 2; ++j)
#pragma unroll
            for (int r = 0; r < 8; ++r) {
                const int o = tileM + 16 * i + r + 8 * half;
                const size_t idx = ((size_t)b * C_SZ + o) * L_SZ + tileN + 16 * j + n;
                const float g = 1.f / (1.f + __expf(-acc[i][j][r]));
                out[idx] = X[idx] * g;
            }
}

// ---------------------------------------------------------------------------
// Launch. Workspace layout (floats):
//   DT: [0,        1M)     4 MB   DCT basis
//   F : [1M,      17M)    64 MB   freq / layernormed h (in place)
//   H1: [17M,     49M)   128 MB   relu(fc1) activations
// ---------------------------------------------------------------------------
extern "C" void kernel_launch(void* const* d_in, const int* in_sizes, int n_in,
                              void* d_out, int out_size, void* d_ws, size_t ws_size,
                              hipStream_t stream) {
    (void)in_sizes; (void)n_in; (void)out_size; (void)ws_size;
    const float* x   = (const float*)d_in[0];
    const float* w1  = (const float*)d_in[1];
    const float* w2  = (const float*)d_in[2];
    const float* lnw = (const float*)d_in[3];
    const float* lnb = (const float*)d_in[4];
    float* out = (float*)d_out;

    float* DT = (float*)d_ws;
    float* F  = DT + (size_t)L_SZ * L_SZ;             // + 1M floats
    float* H1 = F + (size_t)ROWS * L_SZ;              // + 16M floats

    dct_gen_kernel<<<(L_SZ * L_SZ) / 256, 256, 0, stream>>>(DT);
    dct_gemm_kernel<<<(ROWS / 128) * (L_SZ / 128), 256, 0, stream>>>(x, DT, F);
    layernorm_kernel<<<ROWS, 256, 0, stream>>>(F, lnw, lnb);
    fc1_relu_kernel<<<dim3((TWO_C / 128) * (L_SZ / 128), B_SZ), 256, 0, stream>>>(w1, F, H1);
    fc2_gate_kernel<<<dim3((C_SZ / 128) * (L_SZ / 128), B_SZ), 256, 0, stream>>>(w2, H1, x, out);
}